// SAGEinitConv_68281390072361
// MI455X (gfx1250) — compile-verified
//
#include <hip/hip_runtime.h>

// ---------------------------------------------------------------------------
// GraphSAGE conv for MI455X (gfx1250, wave32).
//   k1: zero agg-sum + counts in workspace
//   k2: edge-parallel scatter (coalesced float4 gather + global_atomic_add_f32)
//   k3: fused mean + concat + GEMM via V_WMMA_F32_16X16X4_F32 (native fp32 WMMA)
// N=50000, E=800000, D_IN=128, D_OUT=128, K=256.
// ---------------------------------------------------------------------------

typedef __attribute__((ext_vector_type(2))) float v2f;
typedef __attribute__((ext_vector_type(4))) float v4f;
typedef __attribute__((ext_vector_type(8))) float v8f;

#define D_IN 128
#define D_K 256
#define LDS_STRIDE 260   // 256 + 4 pad: rows land 4 LDS banks apart -> conflict-free A reads

__global__ __launch_bounds__(256) void sage_zero_ws(float* __restrict__ agg,
                                                    float* __restrict__ cnt,
                                                    int nAgg, int nCnt) {
  int i = blockIdx.x * blockDim.x + threadIdx.x;
  int stride = gridDim.x * blockDim.x;
  for (int j = i; j < nAgg; j += stride) agg[j] = 0.0f;
  for (int j = i; j < nCnt; j += stride) cnt[j] = 0.0f;
}

// One wave (32 lanes) per edge; lane l moves features [4l, 4l+4).
__global__ __launch_bounds__(256) void sage_scatter(const float* __restrict__ x,
                                                    const int* __restrict__ ei,
                                                    float* __restrict__ agg,
                                                    float* __restrict__ cnt,
                                                    int E) {
  int gid  = blockIdx.x * blockDim.x + threadIdx.x;
  int e    = gid >> 5;
  int lane = gid & 31;
  if (e >= E) return;
  int src = ei[e];       // row 0: message source
  int dst = ei[E + e];   // row 1: aggregation target
  const v4f xv = *(const v4f*)(x + (size_t)src * D_IN + lane * 4);
  float* a = agg + (size_t)dst * D_IN + lane * 4;
  atomicAdd(a + 0, xv.x);
  atomicAdd(a + 1, xv.y);
  atomicAdd(a + 2, xv.z);
  atomicAdd(a + 3, xv.w);
  if (lane == 0) atomicAdd(cnt + dst, 1.0f);
}

// One workgroup (8 waves) per 16-node tile; wave w owns out cols [16w,16w+16).
__global__ __launch_bounds__(256) void sage_gemm(const float* __restrict__ x,
                                                 const float* __restrict__ w,
                                                 const float* __restrict__ agg,
                                                 const float* __restrict__ cnt,
                                                 float* __restrict__ out,
                                                 int N) {
  __shared__ float hT[16][LDS_STRIDE];

  const int tid  = threadIdx.x;
  const int base = blockIdx.x * 16;

  // --- Stage h = [x | agg/max(cnt,1)] tile (16 x 256) into LDS ---
  // thread t: row = t>>4, 16-col segment = (t&15)*16, via 4x float4.
  {
    const int row  = tid >> 4;
    const int node = base + row;
    const int cseg = (tid & 15) * 16;
    const bool valid = node < N;
    float invc = 1.0f;
    if (valid && cseg >= D_IN) invc = 1.0f / fmaxf(cnt[node], 1.0f);
    #pragma unroll
    for (int j = 0; j < 16; j += 4) {
      const int col = cseg + j;
      v4f v = {0.0f, 0.0f, 0.0f, 0.0f};
      if (valid) {
        if (col < D_IN) {
          v = *(const v4f*)(x + (size_t)node * D_IN + col);
        } else {
          v = *(const v4f*)(agg + (size_t)node * D_IN + (col - D_IN));
          v = v * invc;
        }
      }
      *(v4f*)&hT[row][col] = v;
    }
  }
  __syncthreads();

  // --- WMMA f32 16x16x4, K = 256 -> 64 matrix ops per wave ---
  const int wave = tid >> 5;
  const int lane = tid & 31;
  const int m    = lane & 15;   // A row / B,C,D column-in-tile
  const int hi   = lane >> 4;   // half-wave selects K pair {0,1} vs {2,3}
  const int col  = wave * 16 + m;

  const float* wc = w + col;    // W is [256][128] row-major
  v8f acc = {};
  #pragma unroll 4
  for (int k = 0; k < D_K; k += 4) {
    const int ka = k + 2 * hi;
    v2f a = *(const v2f*)&hT[m][ka];            // ds_load_b64, bank-conflict-free
    v2f b;
    b.x = wc[(size_t)ka * 128];
    b.y = wc[(size_t)(ka + 1) * 128];
    acc = __builtin_amdgcn_wmma_f32_16x16x4_f32(
        /*neg_a=*/false, a, /*neg_b=*/false, b,
        /*c_mod=*/(short)0, acc, /*reuse_a=*/false, /*reuse_b=*/false);
  }

  // --- Store: VGPR v of lane half hi holds row (v + 8*hi), column col ---
  if (base + 15 < N) {
    float* o = out + (size_t)(base + 8 * hi) * 128 + col;
    #pragma unroll
    for (int v = 0; v < 8; ++v) o[(size_t)v * 128] = acc[v];
  } else {
    #pragma unroll
    for (int v = 0; v < 8; ++v) {
      const int r = base + 8 * hi + v;
      if (r < N) out[(size_t)r * 128 + col] = acc[v];
    }
  }
}

extern "C" void kernel_launch(void* const* d_in, const int* in_sizes, int n_in,
                              void* d_out, int out_size, void* d_ws, size_t ws_size,
                              hipStream_t stream) {
  const float* x  = (const float*)d_in[0];   // [N,128] f32
  const int*   ei = (const int*)d_in[1];     // [2,E]
  const float* w  = (const float*)d_in[2];   // [256,128] f32
  float* out = (float*)d_out;                // [N,128] f32

  const int N = in_sizes[0] / D_IN;
  const int E = in_sizes[1] / 2;

  float* agg = (float*)d_ws;                 // [N,128]
  float* cnt = agg + (size_t)N * D_IN;       // [N]

  sage_zero_ws<<<4096, 256, 0, stream>>>(agg, cnt, N * D_IN, N);

  const long long scatterThreads = (long long)E * 32;
  const int scatterBlocks = (int)((scatterThreads + 255) / 256);
  sage_scatter<<<scatterBlocks, 256, 0, stream>>>(x, ei, agg, cnt, E);

  sage_gemm<<<(N + 15) / 16, 256, 0, stream>>>(x, w, agg, cnt, out, N);
}